// MultiHead_71829033059022
// MI455X (gfx1250) — compile-verified
//
#include <hip/hip_runtime.h>
#include <hip/hip_bf16.h>

// Problem constants (fixed by the reference)
#define B_  8
#define T_  12
#define N_  307
#define DM  64          // d_model
#define NPAD 320        // N padded to 20*16 for WMMA tiling; stride 320 floats
#define NTILES 20       // ceil(307/16)

typedef __attribute__((ext_vector_type(2))) float v2f;
typedef __attribute__((ext_vector_type(8))) float v8f;

// ---------------------------------------------------------------------------
// Kernel 1: fused Q/K projection + S = (Q K^T)/sqrt(8) with mask, per (b,t).
// One block per (b,t) slice; 256 threads = 8 wave32.
// LDS holds Q and K feature-major: Qs[e][n], Ks[e][m]  (64 x 320 each).
// Each wave owns whole row-tiles: the A (Q) fragment is loaded into registers
// once per row tile and reused across all 20 column tiles, so the inner loop
// issues only B-side LDS loads between WMMAs.
// ---------------------------------------------------------------------------
__global__ __launch_bounds__(256) void qk_scores_kernel(
    const float* __restrict__ x,      // [B, DM, T, N]
    const int*   __restrict__ mask,   // [B, N, N]
    const float* __restrict__ Wq,     // [DM, DM]
    const float* __restrict__ bq,     // [DM]
    const float* __restrict__ Wk,     // [DM, DM]
    const float* __restrict__ bk,     // [DM]
    float*       __restrict__ out)    // [B, T, N, N] (pre-softmax scores)
{
    __shared__ float Qs[DM * NPAD];   // 80 KB
    __shared__ float Ks[DM * NPAD];   // 80 KB  (160 KB of the 320 KB WGP LDS)

    const int bt = blockIdx.x;
    const int b  = bt / T_;
    const int t  = bt % T_;
    const int tid = threadIdx.x;

    // ---- Phase 1: projections  Q[e][n] = sum_c x[b,c,t,n] * Wq[e,c] + bq[e]
    // Consecutive threads -> consecutive n => coalesced x loads; the 78 KB x
    // slice stays hot in cache across the 128 (e, which) reuses.
    const int TOT = 2 * DM * N_;
    for (int idx = tid; idx < TOT; idx += 256) {
        const int which = (idx >= DM * N_);           // 0 = Q, 1 = K
        const int r = which ? idx - DM * N_ : idx;
        const int e = r / N_;
        const int n = r - e * N_;
        const float* __restrict__ W = which ? Wk : Wq;
        float acc = which ? bk[e] : bq[e];
        const int xbase = ((b * DM) * T_ + t) * N_ + n;   // c = 0 term
        #pragma unroll 8
        for (int c = 0; c < DM; ++c) {
            acc = fmaf(x[xbase + c * (T_ * N_)], W[e * DM + c], acc);
        }
        float* dst = which ? Ks : Qs;
        dst[e * NPAD + n] = acc;
    }
    __syncthreads();

    // ---- Phase 2: S[n][m] = sum_e Qs[e][n] * Ks[e][m] via V_WMMA_F32_16X16X4_F32
    const int wave = tid >> 5;
    const int lane = tid & 31;
    const int half = lane >> 4;       // 0: lanes 0-15, 1: lanes 16-31
    const int l    = lane & 15;
    const float scale = 0.3535533905932738f;  // 1/sqrt(DK), DK = 8

    // A (16x4 f32): lanes 0-15 hold K=0 (x) / K=1 (y) for M=lane,
    //               lanes 16-31 hold K=2 / K=3 for M=lane-16.
    // B (4x16 f32): symmetric — lanes 0-15 rows K=0/1, lanes 16-31 rows K=2/3.
    for (int ti = wave; ti < NTILES; ti += 8) {        // uniform per wave
        const int n0 = ti * 16;

        // Load the Q (A-matrix) fragments for this row tile once: 16 x v2f.
        v2f afrag[16];
        #pragma unroll
        for (int s = 0; s < 16; ++s) {
            const int ka = 4 * s + 2 * half;
            afrag[s].x = Qs[(ka + 0) * NPAD + n0 + l];
            afrag[s].y = Qs[(ka + 1) * NPAD + n0 + l];
        }

        for (int tj = 0; tj < NTILES; ++tj) {
            const int m0 = tj * 16;
            v8f acc = {};
            #pragma unroll
            for (int s = 0; s < 16; ++s) {
                const int ka = 4 * s + 2 * half;
                v2f bb;
                bb.x = Ks[(ka + 0) * NPAD + m0 + l];
                bb.y = Ks[(ka + 1) * NPAD + m0 + l];
                acc = __builtin_amdgcn_wmma_f32_16x16x4_f32(
                    false, afrag[s], false, bb, (short)0, acc, false, false);
            }

            // C/D layout: VGPR v holds row M = v + 8*half, column N = lane & 15.
            const int gm = m0 + l;
            if (gm < N_) {
                #pragma unroll
                for (int v = 0; v < 8; ++v) {
                    const int gn = n0 + v + 8 * half;
                    if (gn < N_) {
                        float val = acc[v] * scale;
                        if (mask[(b * N_ + gn) * N_ + gm] == 0) val = -1.0e9f;
                        out[((size_t)bt * N_ + gn) * N_ + gm] = val;
                    }
                }
            }
        }
    }
}

// ---------------------------------------------------------------------------
// Kernel 2: in-place row softmax over the last axis (row length 307).
// One wave32 per row; 8 rows per 256-thread block. The 36 MB intermediate
// stays resident in the 192 MB L2 between the two kernels.
// ---------------------------------------------------------------------------
#define ROWS_TOTAL (B_ * T_ * N_)
#define ELEMS_PER_LANE 10   // ceil(307/32)

__global__ __launch_bounds__(256) void softmax_rows_kernel(float* __restrict__ out)
{
    const int row  = blockIdx.x * 8 + (threadIdx.x >> 5);
    const int lane = threadIdx.x & 31;
    if (row >= ROWS_TOTAL) return;

    float* __restrict__ p = out + (size_t)row * N_;

    // max
    float m = -3.4e38f;
    for (int j = lane; j < N_; j += 32) m = fmaxf(m, p[j]);
    #pragma unroll
    for (int off = 16; off > 0; off >>= 1) m = fmaxf(m, __shfl_xor(m, off, 32));

    // exp + sum (keep exps in registers)
    float ev[ELEMS_PER_LANE];
    float s = 0.0f;
    int c = 0;
    for (int j = lane; j < N_; j += 32) {
        float e = __expf(p[j] - m);
        ev[c++] = e;
        s += e;
    }
    #pragma unroll
    for (int off = 16; off > 0; off >>= 1) s += __shfl_xor(s, off, 32);

    const float inv = 1.0f / s;
    c = 0;
    for (int j = lane; j < N_; j += 32) p[j] = ev[c++] * inv;
}

// ---------------------------------------------------------------------------
extern "C" void kernel_launch(void* const* d_in, const int* in_sizes, int n_in,
                              void* d_out, int out_size, void* d_ws, size_t ws_size,
                              hipStream_t stream) {
    const float* x    = (const float*)d_in[0];
    const int*   mask = (const int*)  d_in[1];
    const float* Wq   = (const float*)d_in[2];
    const float* bq   = (const float*)d_in[3];
    const float* Wk   = (const float*)d_in[4];
    const float* bk   = (const float*)d_in[5];
    float* out = (float*)d_out;

    qk_scores_kernel<<<B_ * T_, 256, 0, stream>>>(x, mask, Wq, bq, Wk, bk, out);

    const int rows_blocks = (ROWS_TOTAL + 7) / 8;
    softmax_rows_kernel<<<rows_blocks, 256, 0, stream>>>(out);
}